// RealAgnosticDensityResidualInteractionBlock_28939489641129
// MI455X (gfx1250) — compile-verified
//
#include <hip/hip_runtime.h>

#define NN 10000
#define EE 128000
#define CC 128
#define AA 10
#define RR 8

typedef __attribute__((ext_vector_type(2))) float v2f;
typedef __attribute__((ext_vector_type(8))) float v8f;
typedef __attribute__((ext_vector_type(4))) unsigned int v4u;
typedef __attribute__((ext_vector_type(4))) int v4i;
typedef __attribute__((ext_vector_type(8))) int v8i;

__device__ __forceinline__ v8f wmma4(v2f a, v2f b, v8f c) {
  // D = A(16x4) * B(4x16) + C, full f32 precision on the matrix pipe
  return __builtin_amdgcn_wmma_f32_16x16x4_f32(false, a, false, b, (short)0, c, false, false);
}

__device__ __forceinline__ float silu_f(float x) { return x / (1.0f + __expf(-x)); }

// ---------------------------------------------------------------------------
// Tensor Data Mover: 1-D contiguous f32 tile (nelem elements) global -> LDS.
// D# encoding per CDNA5 ISA 8.3/8.4: group0 = {flags, lds_addr, gaddr_lo,
// gaddr_hi|type=2}; group1 = {data_size=4B, tensor_dim0, tensor_dim1=1,
// tile_dim0=nelem, tile_dim1=1, stride0=nelem}. Issue from ONE wave
// (EXEC ignored by TDM), wait TENSORcnt, then block barrier publishes LDS.
// ---------------------------------------------------------------------------
#if defined(__has_builtin)
#if __has_builtin(__builtin_amdgcn_tensor_load_to_lds)
#define HAVE_TDM 1
#endif
#endif

__device__ __forceinline__ void tdm_load_1d_f32(unsigned lds_addr,
                                                const void* gptr,
                                                unsigned nelem) {
#ifdef HAVE_TDM
  unsigned long long ga = (unsigned long long)gptr;
  v4u g0;
  g0[0] = 1u;                                         // count=1, user mode
  g0[1] = lds_addr;                                   // LDS byte address
  g0[2] = (unsigned)(ga & 0xffffffffull);             // global_addr[31:0]
  g0[3] = (unsigned)((ga >> 32) & 0x01ffffffull) | (2u << 30);  // [56:32]|type=2
  v8i g1;
  g1[0] = (int)(2u << 16);            // wg_mask=0, data_size=2 -> 4 bytes
  g1[1] = (int)(nelem << 16);         // tensor_dim0[15:0] (bits 63:48)
  g1[2] = (int)((nelem >> 16) | (1u << 16));  // tensor_dim0 hi | tensor_dim1=1
  g1[3] = (int)(nelem << 16);         // tile_dim0 = nelem (bits 127:112)
  g1[4] = 1;                          // tile_dim1 = 1, tile_dim2 = 0
  g1[5] = (int)nelem;                 // tensor_dim0_stride[31:0]
  g1[6] = 0;
  g1[7] = 0;
  v4i gz = {0, 0, 0, 0};
#if __clang_major__ >= 23
  v8i gz8 = {0, 0, 0, 0, 0, 0, 0, 0};
  __builtin_amdgcn_tensor_load_to_lds(g0, g1, gz, gz, gz8, 0);
#else
  __builtin_amdgcn_tensor_load_to_lds(g0, g1, gz, gz, 0);
#endif
  __builtin_amdgcn_s_wait_tensorcnt(0);
#endif
}

// ---------------------------------------------------------------------------
// Kernel 1: skip connection  sc = [x0*attrs]·W_skip0 , [x1k*attrs]·W_skip1
// X tile (16 x 1280) built once in LDS; inner loop is ds_load_b64 + wmma.
// grid (N/16, 4 comps), block 256 (8 waves = 8 column tiles of C=128)
// ---------------------------------------------------------------------------
__global__ __launch_bounds__(256) void skip_kernel(
    const float* __restrict__ node_attrs, const float* __restrict__ node_feats,
    const float* __restrict__ Wsk0, const float* __restrict__ Wsk1,
    float* __restrict__ out) {
  __shared__ float X[16 * AA * CC];   // 80 KB
  __shared__ float xc[16 * CC];
  __shared__ float at[16 * AA];
  const int comp = blockIdx.y;
  const int n0 = blockIdx.x * 16;
  const int tid = threadIdx.x;

  for (int i = tid; i < 16 * CC; i += 256) {
    int r = i >> 7, u = i & 127;
    int n = n0 + r;
    xc[i] = (comp == 0) ? node_feats[n * (4 * CC) + u]
                        : node_feats[n * (4 * CC) + CC + u * 3 + (comp - 1)];
  }
  for (int i = tid; i < 16 * AA; i += 256) {
    int r = i / AA, a = i - r * AA;
    at[i] = node_attrs[(n0 + r) * AA + a];
  }
  __syncthreads();
  for (int i = tid; i < 16 * AA * CC; i += 256) {
    int r = i / (AA * CC);
    int ua = i - r * (AA * CC);
    int u = ua / AA, a = ua - u * AA;
    X[i] = xc[r * CC + u] * at[r * AA + a];
  }
  __syncthreads();

  const float* W = (comp == 0) ? Wsk0 : Wsk1;  // (C,A,C) flattens to (1280,128)
  const int w = tid >> 5, lane = tid & 31;
  const int half = lane >> 4, l16 = lane & 15;
  const int col = w * 16 + l16;

  v8f acc = {};
  for (int kk = 0; kk < CC * AA; kk += 4) {
    int ka = kk + 2 * half;
    v2f a, b;
    a.x = X[l16 * (AA * CC) + ka];
    a.y = X[l16 * (AA * CC) + ka + 1];
    b.x = W[ka * CC + col];
    b.y = W[(ka + 1) * CC + col];
    acc = wmma4(a, b, acc);
  }

  const float norm = 0.027950849718747374f;  // 1/sqrt(C*A)
  const int scOff = NN * CC * 4;
#pragma unroll
  for (int j = 0; j < 8; ++j) {
    int n = n0 + j + 8 * half;
    float v = acc[j] * norm;
    if (comp == 0) out[scOff + n * 512 + col] = v;
    else           out[scOff + n * 512 + CC + col * 3 + (comp - 1)] = v;
  }
}

// ---------------------------------------------------------------------------
// Kernel 2: up-projection  h0 = x0·W_up0/sqrt(C), h1k = x1k·W_up1/sqrt(C)
// 32 KB node-feature tile staged by the Tensor Data Mover (wave 0 only).
// ---------------------------------------------------------------------------
__global__ __launch_bounds__(256) void up_kernel(
    const float* __restrict__ node_feats,
    const float* __restrict__ Wup0, const float* __restrict__ Wup1,
    float* __restrict__ h) {
  __shared__ float xl[16 * 4 * CC];   // 32 KB
  const int n0 = blockIdx.x * 16;
  const int tid = threadIdx.x;
#ifdef HAVE_TDM
  if (tid < 32) {
    tdm_load_1d_f32((unsigned)(unsigned long long)(uintptr_t)xl,
                    node_feats + (size_t)n0 * 4 * CC, 16 * 4 * CC);
  }
#else
  for (int i = tid; i < 16 * 4 * CC; i += 256)
    xl[i] = node_feats[n0 * 4 * CC + i];
#endif
  __syncthreads();

  const int w = tid >> 5, lane = tid & 31;
  const int half = lane >> 4, l16 = lane & 15;
  const int col = w * 16 + l16;
  const float norm = 0.08838834764831845f;  // 1/sqrt(128)

  for (int comp = 0; comp < 4; ++comp) {
    const float* W = (comp == 0) ? Wup0 : Wup1;
    v8f acc = {};
    for (int kk = 0; kk < CC; kk += 4) {
      int ka = kk + 2 * half;
      v2f a, b;
      if (comp == 0) {
        a.x = xl[l16 * 512 + ka];
        a.y = xl[l16 * 512 + ka + 1];
      } else {
        a.x = xl[l16 * 512 + CC + ka * 3 + (comp - 1)];
        a.y = xl[l16 * 512 + CC + (ka + 1) * 3 + (comp - 1)];
      }
      b.x = W[ka * CC + col];
      b.y = W[(ka + 1) * CC + col];
      acc = wmma4(a, b, acc);
    }
#pragma unroll
    for (int j = 0; j < 8; ++j) {
      int n = n0 + j + 8 * half;
      float v = acc[j] * norm;
      if (comp == 0) h[n * 512 + col] = v;
      else           h[n * 512 + CC + col * 3 + (comp - 1)] = v;
    }
  }
}

// helper: 32x64 @ 64x64 tile GEMM for MLP middle layers (8 tiles / 8 waves)
__device__ __forceinline__ void mlp64(const float* __restrict__ src,
                                      const float* __restrict__ W,
                                      float* __restrict__ dst,
                                      int w, int half, int l16, float s) {
  int mt = w >> 2, nt = w & 3;
  int col = nt * 16 + l16;
  v8f acc = {};
  for (int kk = 0; kk < 64; kk += 4) {
    int ka = kk + 2 * half;
    v2f a, b;
    a.x = src[(mt * 16 + l16) * 64 + ka];
    a.y = src[(mt * 16 + l16) * 64 + ka + 1];
    b.x = W[ka * 64 + col];
    b.y = W[(ka + 1) * 64 + col];
    acc = wmma4(a, b, acc);
  }
#pragma unroll
  for (int j = 0; j < 8; ++j) {
    int row = mt * 16 + j + 8 * half;
    dst[row * 64 + col] = silu_f(acc[j] * s);
  }
}

// ---------------------------------------------------------------------------
// Kernel 3: edge MLP (all layers in LDS via WMMA) + density + fused
// message construction with atomic scatter into M (N, 2C, 4).
// 32 edges / block, 8 waves; edge-feature tile staged by TDM.
// ---------------------------------------------------------------------------
__global__ __launch_bounds__(256) void edge_kernel(
    const float* __restrict__ edge_attrs, const float* __restrict__ edge_feats,
    const int* __restrict__ edge_index,
    const float* __restrict__ R0, const float* __restrict__ R1,
    const float* __restrict__ R2, const float* __restrict__ R3,
    const float* __restrict__ Wd,
    const float* __restrict__ h, float* __restrict__ M,
    float* __restrict__ dens) {
  __shared__ float ef[32 * RR];
  __shared__ float b1[32 * 64];
  __shared__ float b2[32 * 64];
  __shared__ float wb[32 * 512];
  const int e0 = blockIdx.x * 32;
  const int tid = threadIdx.x;

#ifdef HAVE_TDM
  if (tid < 32) {
    tdm_load_1d_f32((unsigned)(unsigned long long)(uintptr_t)ef,
                    edge_feats + (size_t)e0 * RR, 32 * RR);
  }
#else
  ef[tid] = edge_feats[e0 * RR + tid];  // 32*8 == 256
#endif
  __syncthreads();

  const int w = tid >> 5, lane = tid & 31;
  const int half = lane >> 4, l16 = lane & 15;

  // layer 1: ef(32x8) @ R0(8x64) -> silu -> b1
  {
    int mt = w >> 2, nt = w & 3;
    int col = nt * 16 + l16;
    v8f acc = {};
    for (int kk = 0; kk < RR; kk += 4) {
      int ka = kk + 2 * half;
      v2f a, b;
      a.x = ef[(mt * 16 + l16) * RR + ka];
      a.y = ef[(mt * 16 + l16) * RR + ka + 1];
      b.x = R0[ka * 64 + col];
      b.y = R0[(ka + 1) * 64 + col];
      acc = wmma4(a, b, acc);
    }
    const float s = 0.35355339059327373f;  // 1/sqrt(8)
#pragma unroll
    for (int j = 0; j < 8; ++j) {
      int row = mt * 16 + j + 8 * half;
      b1[row * 64 + col] = silu_f(acc[j] * s);
    }
  }
  __syncthreads();
  mlp64(b1, R1, b2, w, half, l16, 0.125f);  // layer 2
  __syncthreads();
  mlp64(b2, R2, b1, w, half, l16, 0.125f);  // layer 3
  __syncthreads();

  // layer 4: b1(32x64) @ R3(64x512) -> wb (no activation)
  for (int t = 0; t < 8; ++t) {
    int id = w + 8 * t;
    int mt = id >> 5, nt = id & 31;
    int col = nt * 16 + l16;
    v8f acc = {};
    for (int kk = 0; kk < 64; kk += 4) {
      int ka = kk + 2 * half;
      v2f a, b;
      a.x = b1[(mt * 16 + l16) * 64 + ka];
      a.y = b1[(mt * 16 + l16) * 64 + ka + 1];
      b.x = R3[ka * 512 + col];
      b.y = R3[(ka + 1) * 512 + col];
      acc = wmma4(a, b, acc);
    }
#pragma unroll
    for (int j = 0; j < 8; ++j) {
      int row = mt * 16 + j + 8 * half;
      wb[row * 512 + col] = acc[j] * 0.125f;
    }
  }
  __syncthreads();

  // density per edge
  if (tid < 32) {
    int e = e0 + tid;
    float d = 0.f;
#pragma unroll
    for (int r = 0; r < RR; ++r) d += ef[tid * RR + r] * Wd[r];
    d *= 0.35355339059327373f;
    atomicAdd(&dens[edge_index[EE + e]], tanhf(d * d));
  }

  // fused message construction + scatter-add into M (N, 2C, 4)
  const float inv_sqrt3 = 0.5773502691896258f;
  for (int idx = tid; idx < 32 * CC; idx += 256) {
    int e = idx >> 7;
    int u = idx & 127;
    int ge = e0 + e;
    int s = edge_index[ge];
    int r = edge_index[EE + ge];
    float w1 = wb[e * 512 + u];
    float w2 = wb[e * 512 + 128 + u];
    float w3 = wb[e * 512 + 256 + u];
    float w4 = wb[e * 512 + 384 + u];
    float hs0 = h[(size_t)s * 512 + u];
    float h1x = h[(size_t)s * 512 + CC + u * 3 + 0];
    float h1y = h[(size_t)s * 512 + CC + u * 3 + 1];
    float h1z = h[(size_t)s * 512 + CC + u * 3 + 2];
    float y0  = edge_attrs[ge * 4 + 0];
    float y1x = edge_attrs[ge * 4 + 1];
    float y1y = edge_attrs[ge * 4 + 2];
    float y1z = edge_attrs[ge * 4 + 3];
    float* Mr = M + (size_t)r * 1024;
    atomicAdd(&Mr[u * 4 + 0], w1 * hs0 * y0);
    atomicAdd(&Mr[u * 4 + 1], w2 * hs0 * y1x);
    atomicAdd(&Mr[u * 4 + 2], w2 * hs0 * y1y);
    atomicAdd(&Mr[u * 4 + 3], w2 * hs0 * y1z);
    float dot = h1x * y1x + h1y * y1y + h1z * y1z;
    atomicAdd(&Mr[(CC + u) * 4 + 0], w4 * dot * inv_sqrt3);
    atomicAdd(&Mr[(CC + u) * 4 + 1], w3 * h1x * y0);
    atomicAdd(&Mr[(CC + u) * 4 + 2], w3 * h1y * y0);
    atomicAdd(&Mr[(CC + u) * 4 + 3], w3 * h1z * y0);
  }
}

// ---------------------------------------------------------------------------
// Kernel 4: out = (M_j · W_lin{0,1}) / (16 * (density+1)) -> message (N,C,4)
// grid (N/16, j=0..3)
// ---------------------------------------------------------------------------
__global__ __launch_bounds__(256) void out_kernel(
    const float* __restrict__ M, const float* __restrict__ dens,
    const float* __restrict__ Wl0, const float* __restrict__ Wl1,
    float* __restrict__ out) {
  __shared__ float Al[16 * 256];
  const int j = blockIdx.y;
  const int n0 = blockIdx.x * 16;
  const int tid = threadIdx.x;
  for (int i = tid; i < 16 * 256; i += 256) {
    int r = i >> 8, u = i & 255;
    Al[i] = M[(size_t)(n0 + r) * 1024 + u * 4 + j];
  }
  __syncthreads();

  const float* W = (j == 0) ? Wl0 : Wl1;
  const int w = tid >> 5, lane = tid & 31;
  const int half = lane >> 4, l16 = lane & 15;
  const int col = w * 16 + l16;

  v8f acc = {};
  for (int kk = 0; kk < 256; kk += 4) {
    if ((kk & 15) == 0 && kk + 32 < 256)
      __builtin_prefetch(W + (kk + 32) * CC + col, 0, 1);  // global_prefetch_b8
    int ka = kk + 2 * half;
    v2f a, b;
    a.x = Al[l16 * 256 + ka];
    a.y = Al[l16 * 256 + ka + 1];
    b.x = W[ka * CC + col];
    b.y = W[(ka + 1) * CC + col];
    acc = wmma4(a, b, acc);
  }
#pragma unroll
  for (int jj = 0; jj < 8; ++jj) {
    int n = n0 + jj + 8 * half;
    float sc = 0.0625f / (dens[n] + 1.0f);  // 1/sqrt(2C)=1/16
    out[(size_t)n * 512 + col * 4 + j] = acc[jj] * sc;
  }
}

extern "C" void kernel_launch(void* const* d_in, const int* in_sizes, int n_in,
                              void* d_out, int out_size, void* d_ws, size_t ws_size,
                              hipStream_t stream) {
  const float* node_attrs = (const float*)d_in[0];
  const float* node_feats = (const float*)d_in[1];
  const float* edge_attrs = (const float*)d_in[2];
  const float* edge_feats = (const float*)d_in[3];
  const int*   edge_index = (const int*)d_in[4];
  const float* W_up0 = (const float*)d_in[5];
  const float* W_up1 = (const float*)d_in[6];
  const float* R0 = (const float*)d_in[7];
  const float* R1 = (const float*)d_in[8];
  const float* R2 = (const float*)d_in[9];
  const float* R3 = (const float*)d_in[10];
  const float* Wd = (const float*)d_in[11];
  const float* Wl0 = (const float*)d_in[12];
  const float* Wl1 = (const float*)d_in[13];
  const float* Wsk0 = (const float*)d_in[14];
  const float* Wsk1 = (const float*)d_in[15];
  float* out = (float*)d_out;

  char* ws = (char*)d_ws;
  float* h    = (float*)ws;                                         // N*512 f32
  float* M    = (float*)(ws + (size_t)NN * 512 * sizeof(float));    // N*1024 f32
  float* dens = (float*)(ws + (size_t)NN * (512 + 1024) * sizeof(float)); // N f32

  // zero accumulators (M and dens are contiguous)
  hipMemsetAsync(M, 0, ((size_t)NN * 1024 + NN) * sizeof(float), stream);

  dim3 blk(256);
  skip_kernel<<<dim3(NN / 16, 4), blk, 0, stream>>>(node_attrs, node_feats,
                                                    Wsk0, Wsk1, out);
  up_kernel<<<dim3(NN / 16), blk, 0, stream>>>(node_feats, W_up0, W_up1, h);
  edge_kernel<<<dim3(EE / 32), blk, 0, stream>>>(edge_attrs, edge_feats,
                                                 edge_index, R0, R1, R2, R3,
                                                 Wd, h, M, dens);
  out_kernel<<<dim3(NN / 16, 4), blk, 0, stream>>>(M, dens, Wl0, Wl1, out);
}